// ODEFunc_56083682951491
// MI455X (gfx1250) — compile-verified
//
#include <hip/hip_runtime.h>
#include <hip/hip_bf16.h>

// Problem constants (from reference): N=100000 nodes, D=128, H=64.
#define NN 100000
#define DD 128
#define HH 64

typedef __attribute__((ext_vector_type(2))) float v2f;
typedef __attribute__((ext_vector_type(8))) float v8f;

// ---------------------------------------------------------------------------
// Kernel 0: zero two N*D float buffers (ws ax-buffer and d_out accumulator).
// ---------------------------------------------------------------------------
__global__ void zero2_kernel(float4* __restrict__ a, float4* __restrict__ b, int n4) {
    int i = blockIdx.x * blockDim.x + threadIdx.x;
    if (i < n4) {
        float4 z = make_float4(0.f, 0.f, 0.f, 0.f);
        a[i] = z;
        b[i] = z;
    }
}

// ---------------------------------------------------------------------------
// Kernel 1: gate = clamp(sigmoid( clip(x@W1^T + b1, +-10) @ W2^T + b2 )).
// One wave handles 16 nodes. hidden computed via V_WMMA_F32_16X16X4_F32:
//   A (16x4 f32):  lane l holds A[m=l&15][k0], A[m][k0+1],  k0 = 2*(l>>4)
//   B (4x16 f32):  lane l holds B[k0][n=l&15], B[k0+1][n]
//   C/D (16x16):   VGPR r: lanes 0-15 -> M=r, lanes 16-31 -> M=r+8, N=lane&15
// 4 H-tiles (H=64) x 32 K-steps (D=128) = 128 WMMA ops per wave.
// ---------------------------------------------------------------------------
__global__ void gate_kernel(const float* __restrict__ x,
                            const float* __restrict__ W1,
                            const float* __restrict__ b1,
                            const float* __restrict__ W2,
                            const float* __restrict__ b2,
                            float* __restrict__ alph) {
    const int wid  = (blockIdx.x * blockDim.x + threadIdx.x) >> 5; // wave id (uniform)
    const int lane = threadIdx.x & 31;
    const int m0   = wid * 16;
    if (m0 >= NN) return;                 // wave-uniform exit: EXEC stays all-ones

    const int half = lane >> 4;           // 0 or 1 -> which K pair
    const int lm   = lane & 15;

    int mrow = m0 + lm;
    if (mrow >= NN) mrow = NN - 1;        // clamp loads; store is masked below

    const float* arow  = x  + (size_t)mrow * DD + 2 * half;
    const float* brow0 = W1 + (size_t)(0 * 16 + lm) * DD + 2 * half;
    const float* brow1 = W1 + (size_t)(1 * 16 + lm) * DD + 2 * half;
    const float* brow2 = W1 + (size_t)(2 * 16 + lm) * DD + 2 * half;
    const float* brow3 = W1 + (size_t)(3 * 16 + lm) * DD + 2 * half;

    v8f acc0 = {}, acc1 = {}, acc2 = {}, acc3 = {};

    #pragma unroll 4
    for (int k = 0; k < DD; k += 4) {
        v2f a   = *(const v2f*)(arow  + k);
        v2f bb0 = *(const v2f*)(brow0 + k);
        v2f bb1 = *(const v2f*)(brow1 + k);
        v2f bb2 = *(const v2f*)(brow2 + k);
        v2f bb3 = *(const v2f*)(brow3 + k);
        acc0 = __builtin_amdgcn_wmma_f32_16x16x4_f32(false, a, false, bb0, (short)0, acc0, false, false);
        acc1 = __builtin_amdgcn_wmma_f32_16x16x4_f32(false, a, false, bb1, (short)0, acc1, false, false);
        acc2 = __builtin_amdgcn_wmma_f32_16x16x4_f32(false, a, false, bb2, (short)0, acc2, false, false);
        acc3 = __builtin_amdgcn_wmma_f32_16x16x4_f32(false, a, false, bb3, (short)0, acc3, false, false);
    }

    // Per-lane column h = tile*16 + lm
    float b1v0 = b1[ 0 + lm], b1v1 = b1[16 + lm], b1v2 = b1[32 + lm], b1v3 = b1[48 + lm];
    float w20  = W2[ 0 + lm], w21  = W2[16 + lm], w22  = W2[32 + lm], w23  = W2[48 + lm];

    float p[8];
    #pragma unroll
    for (int r = 0; r < 8; ++r) {
        float h0 = fminf(fmaxf(acc0[r] + b1v0, -10.f), 10.f);
        float h1 = fminf(fmaxf(acc1[r] + b1v1, -10.f), 10.f);
        float h2 = fminf(fmaxf(acc2[r] + b1v2, -10.f), 10.f);
        float h3 = fminf(fmaxf(acc3[r] + b1v3, -10.f), 10.f);
        p[r] = h0 * w20 + h1 * w21 + h2 * w22 + h3 * w23;
    }

    // Reduce across the 16 lanes of each half-wave (xor masks 1,2,4,8 stay
    // within lanes [0,15] / [16,31]).
    #pragma unroll
    for (int r = 0; r < 8; ++r) {
        float v = p[r];
        v += __shfl_xor(v, 1, 32);
        v += __shfl_xor(v, 2, 32);
        v += __shfl_xor(v, 4, 32);
        v += __shfl_xor(v, 8, 32);
        p[r] = v;
    }

    if (lm == 0) {  // lane 0 writes rows 0..7, lane 16 writes rows 8..15
        float b2s = b2[0];
        #pragma unroll
        for (int r = 0; r < 8; ++r) {
            int m = m0 + r + 8 * half;
            if (m < NN) {
                float z = p[r] + b2s;
                float s = 1.0f / (1.0f + expf(-z));
                s = fminf(fmaxf(s, 1e-6f), 1.0f - 1e-6f);
                alph[m] = s;
            }
        }
    }
}

// ---------------------------------------------------------------------------
// Kernel 2/3: COO SpMM scatter: dst[rows[e]] += vals[e] * src[cols[e]].
// One wave per edge; lane l handles the float4 at d = 4*l (D=128 = 32 lanes
// x 4 floats). Source row load is one fully coalesced 512B burst; the scatter
// uses global_atomic_add_f32 resolved in L2 (x / ax stay L2-resident: 51MB
// each vs 192MB L2).
// ---------------------------------------------------------------------------
__global__ void spmm_kernel(const int* __restrict__ rows,
                            const int* __restrict__ cols,
                            const float* __restrict__ vals,
                            const float* __restrict__ src,
                            float* __restrict__ dst,
                            int nnz) {
    const int e    = (blockIdx.x * blockDim.x + threadIdx.x) >> 5;
    const int lane = threadIdx.x & 31;
    if (e >= nnz) return;

    const int   r = rows[e];
    const int   c = cols[e];
    const float v = vals[e];

    float4 xv = ((const float4*)(src + (size_t)c * DD))[lane];
    float* d  = dst + (size_t)r * DD + lane * 4;
    atomicAdd(d + 0, v * xv.x);
    atomicAdd(d + 1, v * xv.y);
    atomicAdd(d + 2, v * xv.z);
    atomicAdd(d + 3, v * xv.w);
}

// ---------------------------------------------------------------------------
// Kernel 4: out = alph[n] * out - x   (out currently holds A@(A@x)).
// One thread per float4; row = tid>>5, quad = tid&31.
// ---------------------------------------------------------------------------
__global__ void finalize_kernel(const float* __restrict__ x,
                                const float* __restrict__ alph,
                                float* __restrict__ out) {
    int idx  = blockIdx.x * blockDim.x + threadIdx.x;
    int node = idx >> 5;
    int q    = idx & 31;
    if (node >= NN) return;

    float a = alph[node];
    float4*       op = (float4*)(out + (size_t)node * DD) + q;
    const float4* xp = (const float4*)(x + (size_t)node * DD) + q;
    float4 o  = *op;
    float4 xv = *xp;
    o.x = a * o.x - xv.x;
    o.y = a * o.y - xv.y;
    o.z = a * o.z - xv.z;
    o.w = a * o.w - xv.w;
    *op = o;
}

// ---------------------------------------------------------------------------
// Launcher. Inputs (setup_inputs order):
//   0:t  1:x[N*D]  2:rows[NNZ]  3:cols[NNZ]  4:vals[NNZ]
//   5:W1[H*D]  6:b1[H]  7:W2[1*H]  8:b2[1]
// Workspace layout: [0, N*D) floats = ax buffer; [N*D, N*D+N) = alph.
// ---------------------------------------------------------------------------
extern "C" void kernel_launch(void* const* d_in, const int* in_sizes, int n_in,
                              void* d_out, int out_size, void* d_ws, size_t ws_size,
                              hipStream_t stream) {
    const float* x    = (const float*)d_in[1];
    const int*   rows = (const int*)  d_in[2];
    const int*   cols = (const int*)  d_in[3];
    const float* vals = (const float*)d_in[4];
    const float* W1   = (const float*)d_in[5];
    const float* b1   = (const float*)d_in[6];
    const float* W2   = (const float*)d_in[7];
    const float* b2   = (const float*)d_in[8];
    const int    nnz  = in_sizes[2];

    float* out   = (float*)d_out;
    float* ws_ax = (float*)d_ws;
    float* alph  = ws_ax + (size_t)NN * DD;

    // 0) zero accumulators (ws and d_out are poisoned by the harness)
    {
        int n4 = NN * DD / 4;
        int blocks = (n4 + 255) / 256;
        zero2_kernel<<<blocks, 256, 0, stream>>>((float4*)ws_ax, (float4*)out, n4);
    }

    // 1) gate -> alph  (one wave per 16 nodes, 8 waves / 256-thread block)
    {
        int waves  = (NN + 15) / 16;           // 6250
        int blocks = (waves + 7) / 8;          // 782
        gate_kernel<<<blocks, 256, 0, stream>>>(x, W1, b1, W2, b2, alph);
    }

    // 2) ax = A @ x   (wave per edge)
    {
        long long threads = (long long)nnz * 32;
        int blocks = (int)((threads + 255) / 256);
        spmm_kernel<<<blocks, 256, 0, stream>>>(rows, cols, vals, x, ws_ax, nnz);
    }

    // 3) out = A @ ax
    {
        long long threads = (long long)nnz * 32;
        int blocks = (int)((threads + 255) / 256);
        spmm_kernel<<<blocks, 256, 0, stream>>>(rows, cols, vals, ws_ax, out, nnz);
    }

    // 4) out = alph * out - x
    {
        int threads = NN * 32;
        int blocks  = (threads + 255) / 256;
        finalize_kernel<<<blocks, 256, 0, stream>>>(x, alph, out);
    }
}